// QModel_65481071409913
// MI455X (gfx1250) — compile-verified
//
#include <hip/hip_runtime.h>
#include <math.h>

typedef __attribute__((ext_vector_type(2))) float v2f;
typedef __attribute__((ext_vector_type(8))) float v8f;

// ---- workspace layout (floats) ----
#define OFF_WCR   0      // 13 used: 2*w_q*Re(c_q)
#define OFF_WCI   16     // 13 used: 2*w_q*Im(c_q)
#define OFF_CMAT  32     // 256: 16x16 contraction matrix for the Gram
#define OFF_BH    288    // 1: bias
#define OFF_DLOW  320    // 8192: per-r diagonal coefficient (wires 3..15)
#define OFF_PA    8512   // 1024: partials from k_pairs  (b*8+u)
#define OFF_PB    9536   // 2048: partials from k_gram   ((b*8+chunk)*2 = {num, trace})

struct cpx { float x, y; };
__device__ inline cpx cmul(cpx a, cpx b){ return {a.x*b.x - a.y*b.y, a.x*b.y + a.y*b.x}; }
__device__ inline cpx cadd(cpx a, cpx b){ return {a.x+b.x, a.y+b.y}; }
__device__ inline cpx csub(cpx a, cpx b){ return {a.x-b.x, a.y-b.y}; }
__device__ inline cpx cconj(cpx a){ return {a.x, -a.y}; }
__device__ inline cpx cscale(cpx a, float s){ return {a.x*s, a.y*s}; }

// ---------------------------------------------------------------------------
// Kernel P: fold the whole circuit into constants.
//   wires 3..15: M_w = U4 U3 U2 U1 (2x2), A = M†ZM -> diag (a0,a1) + offdiag c
//   wires 0..2 : W = Π_l [C12 C01 (U0⊗U1⊗U2)] (8x8), Bc = Σ_q w_q W†Z_qW
//   Cmat(16x16) so that num_B = Σ Cmat ∘ (Y·Yᵀ);  Dlow[r] = Σ_q w_q a_{bit}(r)
// ---------------------------------------------------------------------------
__global__ void __launch_bounds__(256) k_precompute(const float* __restrict__ weights,
                                                    const float* __restrict__ w_head,
                                                    const float* __restrict__ b_head,
                                                    float* __restrict__ ws)
{
    __shared__ float swa0[13], swa1[13];
    if (threadIdx.x == 0) {
        // ---- low wires 3..15: accumulated 2x2 unitary per wire ----
        for (int w = 3; w < 16; ++w) {
            cpx m00={1.f,0.f}, m01={0.f,0.f}, m10={0.f,0.f}, m11={1.f,0.f};
            for (int l = 0; l < 4; ++l) {
                float th = weights[(l*16 + w)*2 + 0];
                float ph = weights[(l*16 + w)*2 + 1];
                float c = cosf(0.5f*th), s = sinf(0.5f*th);
                float pc = cosf(0.5f*ph), ps = sinf(0.5f*ph);
                cpx e0 = {pc, -ps}, e1 = {pc, ps};           // e^{-i p/2}, e^{+i p/2}
                cpx u00 = cscale(e0,  c), u01 = cscale(e0, -s);
                cpx u10 = cscale(e1,  s), u11 = cscale(e1,  c);
                cpx n00 = cadd(cmul(u00,m00), cmul(u01,m10));
                cpx n01 = cadd(cmul(u00,m01), cmul(u01,m11));
                cpx n10 = cadd(cmul(u10,m00), cmul(u11,m10));
                cpx n11 = cadd(cmul(u10,m01), cmul(u11,m11));
                m00=n00; m01=n01; m10=n10; m11=n11;
            }
            float wq = w_head[w];
            float a0 = (m00.x*m00.x + m00.y*m00.y) - (m10.x*m10.x + m10.y*m10.y);
            float a1 = (m01.x*m01.x + m01.y*m01.y) - (m11.x*m11.x + m11.y*m11.y);
            cpx cc = csub(cmul(cconj(m00), m01), cmul(cconj(m10), m11));
            swa0[w-3] = wq*a0;
            swa1[w-3] = wq*a1;
            ws[OFF_WCR + (w-3)] = 2.f*wq*cc.x;
            ws[OFF_WCI + (w-3)] = 2.f*wq*cc.y;
        }
        // ---- top-3-bit 8x8 unitary W (wires 0,1,2 + CNOTs) ----
        cpx W[8][8];
        for (int u=0;u<8;u++) for (int v=0;v<8;v++) W[u][v] = { (u==v)?1.f:0.f, 0.f };
        for (int l = 0; l < 4; ++l) {
            cpx U[3][2][2];
            for (int w = 0; w < 3; ++w) {
                float th = weights[(l*16+w)*2+0], ph = weights[(l*16+w)*2+1];
                float c=cosf(.5f*th), s=sinf(.5f*th), pc=cosf(.5f*ph), ps=sinf(.5f*ph);
                cpx e0={pc,-ps}, e1={pc,ps};
                U[w][0][0]=cscale(e0,c);  U[w][0][1]=cscale(e0,-s);
                U[w][1][0]=cscale(e1,s);  U[w][1][1]=cscale(e1,c);
            }
            cpx T[8][8], R[8][8];
            for (int u = 0; u < 8; ++u) {
                int u1 = u  ^ ((u >> 1) & 1);          // CNOT(1,2): flip bit0 if bit1
                int up = u1 ^ (((u1 >> 2) & 1) << 1);  // CNOT(0,1): flip bit1 if bit2
                for (int v = 0; v < 8; ++v)
                    T[u][v] = cmul(cmul(U[0][(up>>2)&1][(v>>2)&1],
                                        U[1][(up>>1)&1][(v>>1)&1]),
                                        U[2][ up     &1][ v     &1]);
            }
            for (int u=0;u<8;u++) for (int v=0;v<8;v++) {
                cpx s = {0.f,0.f};
                for (int k=0;k<8;k++) s = cadd(s, cmul(T[u][k], W[k][v]));
                R[u][v] = s;
            }
            for (int u=0;u<8;u++) for (int v=0;v<8;v++) W[u][v] = R[u][v];
        }
        float zw[8];
        for (int k=0;k<8;k++)
            zw[k] = w_head[0]*(1.f-2.f*((k>>2)&1))
                  + w_head[1]*(1.f-2.f*((k>>1)&1))
                  + w_head[2]*(1.f-2.f*( k    &1));
        for (int u=0;u<8;u++) for (int v=0;v<8;v++) {
            cpx s = {0.f,0.f};
            for (int k=0;k<8;k++) s = cadd(s, cscale(cmul(cconj(W[k][u]), W[k][v]), zw[k]));
            ws[OFF_CMAT +    u *16 +    v ] =  s.x;   // S on P block
            ws[OFF_CMAT + (8+u)*16 + (8+v)] =  s.x;   // S on Q block
            ws[OFF_CMAT +    u *16 + (8+v)] = -s.y;   // -T on R block
            ws[OFF_CMAT + (8+u)*16 +    v ] =  s.y;   // +T on R^T block
        }
        ws[OFF_BH] = b_head[0];
    }
    __syncthreads();
    for (int r = threadIdx.x; r < 8192; r += 256) {
        float d = 0.f;
        for (int j = 0; j < 13; ++j)
            d += ((r >> (12 - j)) & 1) ? swa1[j] : swa0[j];
        ws[OFF_DLOW + r] = d;
    }
}

// ---------------------------------------------------------------------------
// Kernel B: per (batch, r-chunk) compute 16x16 Gram Y·Yᵀ with V_WMMA_F32_16X16X4_F32
// (A and B operands are the same registers), contract with Cmat, plus Dlow·|ψ|²
// diagonal term, plus trace (= norm²). Writes {num, trace} partials.
// ---------------------------------------------------------------------------
__global__ void __launch_bounds__(256) k_gram(const float* __restrict__ sre_g,
                                              const float* __restrict__ sim_g,
                                              const float* __restrict__ ws,
                                              float* __restrict__ pB)
{
    constexpr int TSTR = 260;                 // padded row stride -> conflict-free
    __shared__ float tile[16*TSTR];           // 16 rows x 256 cols
    __shared__ float gram[8*256];             // per-wave 16x16 partials
    __shared__ float red[256];
    const int tid   = threadIdx.x;
    const int b     = blockIdx.x >> 3;
    const int chunk = blockIdx.x & 7;
    const int lane  = tid & 31;
    const int wave  = tid >> 5;
    const int row   = lane & 15;              // WMMA M (A) / N (B) index
    const int hi    = lane >> 4;              // lanes 16-31 carry K=2,3 / M+8
    v8f acc = {0.f,0.f,0.f,0.f,0.f,0.f,0.f,0.f};
    float accDiag = 0.f;

    const int ldrow = tid >> 4;               // 0..15 (rows 0-7 = re, 8-15 = im)
    const int ldcol = (tid & 15) * 16;
    const float* src = (ldrow < 8 ? sre_g : sim_g)
                     + (size_t)b*65536 + (size_t)(ldrow & 7)*8192;

    for (int t = 0; t < 4; ++t) {
        const int tbase = chunk*1024 + t*256;
        const float* gp = src + tbase + ldcol;
        float* lp = &tile[ldrow*TSTR + ldcol];
        #pragma unroll
        for (int k = 0; k < 4; ++k)
            *(float4*)(lp + 4*k) = *(const float4*)(gp + 4*k);
        __syncthreads();

        // diagonal term: Dlow[r] * Σ_rows Y[row,r]^2   (one column per thread)
        {
            float p = 0.f;
            #pragma unroll
            for (int r = 0; r < 16; ++r) { float v = tile[r*TSTR + tid]; p = fmaf(v, v, p); }
            accDiag = fmaf(ws[OFF_DLOW + tbase + tid], p, accDiag);
        }
        // Gram accumulation: each wave owns 32 columns of this tile (8 K-steps of 4)
        #pragma unroll
        for (int k = 0; k < 8; ++k) {
            const int c0 = wave*32 + k*4 + 2*hi;
            v2f ab = *(const v2f*)&tile[row*TSTR + c0];   // {K0,K1} or {K2,K3}
            acc = __builtin_amdgcn_wmma_f32_16x16x4_f32(
                      false, ab, false, ab, (short)0, acc, false, false);
        }
        __syncthreads();
    }
    // dump per-wave 16x16 D (VGPR g: M=g lanes0-15, M=g+8 lanes16-31; N=lane&15)
    #pragma unroll
    for (int g = 0; g < 8; ++g)
        gram[wave*256 + (g + 8*hi)*16 + row] = acc[g];
    __syncthreads();

    float G = 0.f;
    #pragma unroll
    for (int w = 0; w < 8; ++w) G += gram[w*256 + tid];
    float numv = fmaf(ws[OFF_CMAT + tid], G, accDiag);
    float trv  = ((tid >> 4) == (tid & 15)) ? G : 0.f;   // norm² = trace

    red[tid] = numv; __syncthreads();
    for (int s = 128; s > 0; s >>= 1) { if (tid < s) red[tid] += red[tid+s]; __syncthreads(); }
    float totNum = red[0]; __syncthreads();
    red[tid] = trv; __syncthreads();
    for (int s = 128; s > 0; s >>= 1) { if (tid < s) red[tid] += red[tid+s]; __syncthreads(); }
    if (tid == 0) { pB[blockIdx.x*2+0] = totNum; pB[blockIdx.x*2+1] = red[0]; }
}

// ---------------------------------------------------------------------------
// Kernel A: per (batch, u-block) the 13 strided pair correlations for wires 3..15:
//   Σ_pairs 2 w_q Re(c_q · conj(x0) x1), strides 4096..1, all inside one 64KB u-block.
// ---------------------------------------------------------------------------
__global__ void __launch_bounds__(256) k_pairs(const float* __restrict__ sre_g,
                                               const float* __restrict__ sim_g,
                                               const float* __restrict__ ws,
                                               float* __restrict__ pA)
{
    __shared__ float sre[8192];
    __shared__ float sim[8192];
    const int tid = threadIdx.x;
    const int b = blockIdx.x >> 3;
    const int u = blockIdx.x & 7;
    const size_t base = (size_t)b*65536 + (size_t)u*8192;
    #pragma unroll
    for (int k = 0; k < 8; ++k) {
        int idx = (k*256 + tid)*4;
        *(float4*)&sre[idx] = *(const float4*)&sre_g[base + idx];
        *(float4*)&sim[idx] = *(const float4*)&sim_g[base + idx];
    }
    __syncthreads();
    float acc = 0.f;
    for (int j = 0; j < 13; ++j) {            // wire q = j+3, stride = 2^(12-j)
        const int s    = 4096 >> j;
        const int mask = s - 1;
        const float wcr = ws[OFF_WCR + j];
        const float wci = ws[OFF_WCI + j];
        #pragma unroll 4
        for (int p = tid; p < 4096; p += 256) {
            const int i = ((p & ~mask) << 1) | (p & mask);
            float x0r = sre[i],   x0i = sim[i];
            float x1r = sre[i+s], x1i = sim[i+s];
            float t1 = fmaf(x0r, x1r,  x0i*x1i);     // Re(conj(x0)x1)
            float t2 = fmaf(x0r, x1i, -(x0i*x1r));   // Im(conj(x0)x1)
            acc = fmaf(wcr, t1, acc);
            acc = fmaf(-wci, t2, acc);
        }
    }
    __syncthreads();
    sre[tid] = acc; __syncthreads();
    for (int s2 = 128; s2 > 0; s2 >>= 1) { if (tid < s2) sre[tid] += sre[tid+s2]; __syncthreads(); }
    if (tid == 0) pA[blockIdx.x] = sre[0];
}

// ---------------------------------------------------------------------------
// Kernel F: pred[b] = (Σ partials)/norm² + bias
// ---------------------------------------------------------------------------
__global__ void __launch_bounds__(128) k_final(const float* __restrict__ ws,
                                               float* __restrict__ out)
{
    const int b = threadIdx.x;
    float num = 0.f, tr = 0.f;
    #pragma unroll
    for (int u = 0; u < 8; ++u) num += ws[OFF_PA + b*8 + u];
    #pragma unroll
    for (int c = 0; c < 8; ++c) {
        num += ws[OFF_PB + (b*8 + c)*2 + 0];
        tr  += ws[OFF_PB + (b*8 + c)*2 + 1];
    }
    out[b] = num / tr + ws[OFF_BH];
}

extern "C" void kernel_launch(void* const* d_in, const int* in_sizes, int n_in,
                              void* d_out, int out_size, void* d_ws, size_t ws_size,
                              hipStream_t stream) {
    (void)in_sizes; (void)n_in; (void)out_size; (void)ws_size;
    const float* state_re = (const float*)d_in[0];
    const float* state_im = (const float*)d_in[1];
    const float* weights  = (const float*)d_in[2];
    const float* w_head   = (const float*)d_in[3];
    const float* b_head   = (const float*)d_in[4];
    float* ws  = (float*)d_ws;
    float* out = (float*)d_out;

    k_precompute<<<1,    256, 0, stream>>>(weights, w_head, b_head, ws);
    k_gram      <<<1024, 256, 0, stream>>>(state_re, state_im, ws, ws + OFF_PB);
    k_pairs     <<<1024, 256, 0, stream>>>(state_re, state_im, ws, ws + OFF_PA);
    k_final     <<<1,    128, 0, stream>>>(ws, out);
}